// TypeGraphConvolution_79121887527169
// MI455X (gfx1250) — compile-verified
//
#include <hip/hip_runtime.h>
#include <hip/hip_bf16.h>

// Problem sizes (fixed by the reference)
#define B_  8
#define L_  128
#define D_  256

typedef __attribute__((ext_vector_type(2))) float v2f;
typedef __attribute__((ext_vector_type(8))) float v8f;

// ---------------------------------------------------------------------------
// Stage 1: V[b,i,d] = sum_j adj[b,i,j] * (text[b,j,d] + dep[b,j,i,d])
// One block per (b,i); 64 threads, thread t owns d = 4t..4t+3 (float4).
// Each j reads a contiguous 1 KB dep row via global_load_b128 -> dep is
// streamed exactly once (134 MB total -> HBM roofline ~6 us).
// ---------------------------------------------------------------------------
__global__ __launch_bounds__(64) void tgc_stage1(
    const float* __restrict__ text,   // [B,L,D]
    const float* __restrict__ adj,    // [B,L,L]
    const float* __restrict__ dep,    // [B,L,L,D]
    float* __restrict__ V)            // [B,L,D]
{
    const int bi = blockIdx.x;        // 0..B*L-1
    const int b  = bi >> 7;           // / L_
    const int i  = bi & (L_ - 1);
    const int t  = threadIdx.x;       // 0..63  (d-quad index)

    __shared__ float sAdj[L_];
    {
        const float* arow = adj + ((size_t)b * L_ + i) * L_;
        sAdj[t]      = arow[t];
        sAdj[t + 64] = arow[t + 64];
    }
    __syncthreads();

    const float4* tbase = (const float4*)(text + (size_t)b * L_ * D_) + t;                // +D_/4 per j
    const float4* dbase = (const float4*)(dep + (((size_t)b * L_) * L_ + i) * D_) + t;    // +L_*D_/4 per j
    const size_t dstride = (size_t)L_ * D_ / 4;   // in float4 units

    float4 acc = make_float4(0.f, 0.f, 0.f, 0.f);
#pragma unroll 4
    for (int j = 0; j < L_; ++j) {
        if (j + 8 < L_)
            __builtin_prefetch(dbase + (size_t)(j + 8) * dstride, 0, 1);
        const float  a  = sAdj[j];
        const float4 tv = tbase[(size_t)j * (D_ / 4)];
        const float4 dv = dbase[(size_t)j * dstride];
        acc.x = fmaf(a, tv.x + dv.x, acc.x);
        acc.y = fmaf(a, tv.y + dv.y, acc.y);
        acc.z = fmaf(a, tv.z + dv.z, acc.z);
        acc.w = fmaf(a, tv.w + dv.w, acc.w);
    }
    ((float4*)V)[(size_t)bi * (D_ / 4) + t] = acc;
}

// ---------------------------------------------------------------------------
// Stage 2: out[m, n] = V[m, :] @ W[:, n] + bias[n]   (M=1024, K=256, N=256)
// One wave32 per 16x16 output tile using V_WMMA_F32_16X16X4_F32, K stepped
// by 4. Bias is folded into the C accumulator initialization.
// ---------------------------------------------------------------------------
__global__ __launch_bounds__(128) void tgc_stage2(
    const float* __restrict__ V,      // [M=1024, K=256] row-major
    const float* __restrict__ W,      // [K=256, N=256]  row-major
    const float* __restrict__ bias,   // [N=256]
    float* __restrict__ out)          // [M=1024, N=256]
{
    const int lane = threadIdx.x & 31;
    const int wave = threadIdx.x >> 5;                 // 0..3
    const int tile = blockIdx.x * 4 + wave;            // 0..1023
    const int m0   = (tile >> 4) * 16;                 // 64 M-tiles
    const int n0   = (tile & 15) * 16;                 // 16 N-tiles
    const int ln   = lane & 15;
    const int hi   = lane >> 4;                        // lane group: 0 or 1
    const int koff = hi * 2;                           // K sub-offset per ISA A layout

    // A operand: lanes 0-15 -> row M=ln, K={k,k+1}; lanes 16-31 -> K={k+2,k+3}
    const float* aptr = V + (size_t)(m0 + ln) * D_ + koff;
    // B operand: lanes 0-15 -> col N=ln, rows K={k,k+1}; lanes 16-31 -> K={k+2,k+3}
    const float* bptr = W + (size_t)koff * D_ + (n0 + ln);

    // C/D layout: VGPR v holds (row = m0 + hi*8 + v, col = n0 + ln)
    v8f c;
    const float bv = bias[n0 + ln];
#pragma unroll
    for (int v = 0; v < 8; ++v) c[v] = bv;

#pragma unroll 4
    for (int k = 0; k < D_; k += 4) {
        v2f a = *(const v2f*)(aptr + k);               // 8-byte aligned (koff even)
        v2f bm;
        bm.x = bptr[(size_t)k * D_];
        bm.y = bptr[(size_t)k * D_ + D_];
        // (neg_a, A, neg_b, B, c_mod, C, reuse_a, reuse_b)
        c = __builtin_amdgcn_wmma_f32_16x16x4_f32(
                false, a, false, bm, (short)0, c, false, false);
    }

    const int mrow = m0 + hi * 8;
#pragma unroll
    for (int v = 0; v < 8; ++v)
        out[(size_t)(mrow + v) * D_ + (n0 + ln)] = c[v];
}

// ---------------------------------------------------------------------------
// Inputs (setup_inputs order): text[B,L,D] f32, adj[B,L,L] f32,
// dep_embed[B,L,L,D] f32, W[D,D] f32, b[D] f32. Output: [B,L,D] f32.
// Workspace: V needs B*L*D*4 = 1 MB.
// ---------------------------------------------------------------------------
extern "C" void kernel_launch(void* const* d_in, const int* in_sizes, int n_in,
                              void* d_out, int out_size, void* d_ws, size_t ws_size,
                              hipStream_t stream)
{
    const float* text = (const float*)d_in[0];
    const float* adj  = (const float*)d_in[1];
    const float* dep  = (const float*)d_in[2];
    const float* W    = (const float*)d_in[3];
    const float* bias = (const float*)d_in[4];
    float* out = (float*)d_out;
    float* V   = (float*)d_ws;   // 1 MB scratch for intermediate V[b,i,d]

    (void)in_sizes; (void)n_in; (void)out_size; (void)ws_size;

    // Stage 1: 1024 blocks x 64 threads (2 waves) — streaming b128, bandwidth-bound.
    tgc_stage1<<<B_ * L_, 64, 0, stream>>>(text, adj, dep, V);

    // Stage 2: 1024 16x16 tiles, 4 waves/block -> 256 blocks.
    tgc_stage2<<<(B_ * L_ * D_) / (16 * 16 * 4), 128, 0, stream>>>(V, W, bias, out);
}